// Net_90297392431232
// MI455X (gfx1250) — compile-verified
//
#include <hip/hip_runtime.h>

#define HID 128
#define NLAYERS 11

typedef __attribute__((ext_vector_type(2))) float v2f;
typedef __attribute__((ext_vector_type(8))) float v8f;

// ---------------- utility: zero a float buffer ----------------
__global__ void zero_kernel(float* __restrict__ p, int n) {
    int i = blockIdx.x * blockDim.x + threadIdx.x;
    if (i < n) p[i] = 0.0f;
}

// Pack W (L x 128 x 128, k-major) into WMMA B-fragment order so the GEMM
// loads each lane's (W[k][n], W[k+1][n]) pair as one coalesced b64:
// P (in float2 units)[ l*8192 + (k>>2)*256 + ((k>>1)&1)*128 + n ], component (k&1)
__global__ void pack_weights_kernel(const float* __restrict__ W,
                                    float* __restrict__ P, int total) {
    int gid = blockIdx.x * blockDim.x + threadIdx.x;
    if (gid >= total) return;
    int n = gid & 127;
    int k = (gid >> 7) & 127;
    int l = gid >> 14;
    int k0 = k >> 2, hi = (k >> 1) & 1, r = k & 1;
    P[((size_t)l << 14) + (((k0 << 8) + (hi << 7) + n) << 1) + r] = W[gid];
}

// ---------------- conv_first: scatter x (dim 3) ----------------
__global__ void scatter3_kernel(const float* __restrict__ x,
                                const int* __restrict__ src,
                                const int* __restrict__ dst,
                                float* __restrict__ agg3, int n_edges) {
    int e = blockIdx.x * blockDim.x + threadIdx.x;
    if (e >= n_edges) return;
    int s = src[e], d = dst[e];
#pragma unroll
    for (int k = 0; k < 3; ++k)
        unsafeAtomicAdd(&agg3[d * 3 + k], x[s * 3 + k]);
}

// conv_first GEMM: K=3 is tiny -> plain VALU kernel, one thread per output
__global__ void conv_first_kernel(const float* __restrict__ agg3,
                                  const float* __restrict__ x,
                                  const float* __restrict__ Wr,   // 3 x 128
                                  const float* __restrict__ Wt,   // 3 x 128
                                  const float* __restrict__ b,    // 128
                                  float* __restrict__ h, int n_nodes) {
    int gid = blockIdx.x * blockDim.x + threadIdx.x;
    if (gid >= n_nodes * HID) return;
    int i = gid >> 7, n = gid & 127;
    float acc = b[n];
#pragma unroll
    for (int k = 0; k < 3; ++k)
        acc += agg3[i * 3 + k] * Wr[k * HID + n] + x[i * 3 + k] * Wt[k * HID + n];
    h[gid] = fmaxf(acc, 0.0f);
}

// ---------------- scatter for 128-wide features ----------------
// 32 lanes per edge, each lane handles 4 channels via float4 load + hw fp atomics
__global__ void scatter128_kernel(const float* __restrict__ h,
                                  const int* __restrict__ src,
                                  const int* __restrict__ dst,
                                  float* __restrict__ agg, int n_edges) {
    int gid = blockIdx.x * blockDim.x + threadIdx.x;
    int e = gid >> 5;
    if (e >= n_edges) return;
    int c = (gid & 31) * 4;
    int s = src[e], d = dst[e];
    const float4 v = *(const float4*)(h + (size_t)s * HID + c);
    float* out = agg + (size_t)d * HID + c;
    unsafeAtomicAdd(out + 0, v.x);
    unsafeAtomicAdd(out + 1, v.y);
    unsafeAtomicAdd(out + 2, v.z);
    unsafeAtomicAdd(out + 3, v.w);
}

// ---------------- residual GraphConv layer via f32 WMMA ----------------
// out = relu(agg @ Wr + hin @ Wt + bias) + hin ; then re-zero agg rows.
// Block: 256 threads = 8 waves, owns a 32-row M-tile; wave w owns N-tile
// [16w, 16w+16) for both M halves (B fragment reused across the two WMMAs).
__global__ __launch_bounds__(256)
void gconv_gemm_kernel(float* __restrict__ agg,
                       const float* __restrict__ hin,
                       const float* __restrict__ PWr,   // packed 128x128
                       const float* __restrict__ PWt,   // packed 128x128
                       const float* __restrict__ bias,  // 128
                       float* __restrict__ hout) {
    const int lane  = threadIdx.x & 31;
    const int wave  = threadIdx.x >> 5;
    const int mbase = blockIdx.x * 32;
    const int nbase = wave * 16;
    const int lm    = lane & 15;   // M index (A), N index (B/C)
    const int hi    = lane >> 4;   // half-wave select

    // A layout (16x4 f32): lane 0-15 -> M, VGPR0/1 = K=(2*hi), K=(2*hi+1)
    const float* aP0 = agg + (size_t)(mbase + lm) * HID + 2 * hi;
    const float* aP1 = aP0 + 16 * HID;
    const float* hP0 = hin + (size_t)(mbase + lm) * HID + 2 * hi;
    const float* hP1 = hP0 + 16 * HID;
    // packed B: one b64 per matrix per k-step
    const v2f* bRp = (const v2f*)PWr + (hi << 7) + nbase + lm;
    const v2f* bTp = (const v2f*)PWt + (hi << 7) + nbase + lm;

    v8f c00 = {}, c01 = {};   // agg @ Wr, M halves 0/1
    v8f c10 = {}, c11 = {};   // hin @ Wt, M halves 0/1

#pragma unroll 4
    for (int k0 = 0; k0 < HID; k0 += 4) {
        v2f aA0 = *(const v2f*)(aP0 + k0);
        v2f aA1 = *(const v2f*)(aP1 + k0);
        v2f aH0 = *(const v2f*)(hP0 + k0);
        v2f aH1 = *(const v2f*)(hP1 + k0);
        v2f bR  = bRp[(size_t)(k0 >> 2) << 8];
        v2f bT  = bTp[(size_t)(k0 >> 2) << 8];
        c00 = __builtin_amdgcn_wmma_f32_16x16x4_f32(false, aA0, false, bR,
                                                    (short)0, c00, false, false);
        c01 = __builtin_amdgcn_wmma_f32_16x16x4_f32(false, aA1, false, bR,
                                                    (short)0, c01, false, false);
        c10 = __builtin_amdgcn_wmma_f32_16x16x4_f32(false, aH0, false, bT,
                                                    (short)0, c10, false, false);
        c11 = __builtin_amdgcn_wmma_f32_16x16x4_f32(false, aH1, false, bT,
                                                    (short)0, c11, false, false);
    }

    // All waves have consumed agg rows [mbase, mbase+32): re-zero them so the
    // next layer's scatter finds a clean accumulator (saves a full zero pass).
    __syncthreads();
    {
        float4 z = {0.0f, 0.0f, 0.0f, 0.0f};
        float4* az = (float4*)(agg + (size_t)mbase * HID);  // 1024 float4
#pragma unroll
        for (int t = 0; t < 4; ++t)
            az[threadIdx.x + t * 256] = z;
    }

    // C/D layout: VGPR j -> row (+ j + 8*hi), col (nbase + lm)
    const int col = nbase + lm;
    const float bn = bias[col];
#pragma unroll
    for (int g = 0; g < 2; ++g) {
        const v8f cr = g ? c01 : c00;
        const v8f ct = g ? c11 : c10;
        const size_t rb = (size_t)(mbase + g * 16 + hi * 8) * HID + col;
#pragma unroll
        for (int j = 0; j < 8; ++j) {
            float v = fmaxf(cr[j] + ct[j] + bn, 0.0f) + hin[rb + (size_t)j * HID];
            hout[rb + (size_t)j * HID] = v;
        }
    }
}

// ---------------- conv_pos: 128 -> 3, no relu/residual ----------------
__global__ void conv_pos_kernel(const float* __restrict__ agg,
                                const float* __restrict__ h,
                                const float* __restrict__ Wr,  // 128 x 3
                                const float* __restrict__ Wt,  // 128 x 3
                                const float* __restrict__ b,   // 3
                                float* __restrict__ pos, int n_nodes) {
    int gid = blockIdx.x * blockDim.x + threadIdx.x;
    if (gid >= n_nodes * 3) return;
    int i = gid / 3, n = gid - i * 3;
    float acc = b[n];
    for (int k = 0; k < HID; ++k)
        acc += agg[(size_t)i * HID + k] * Wr[k * 3 + n] +
               h[(size_t)i * HID + k] * Wt[k * 3 + n];
    pos[gid] = acc;
}

extern "C" void kernel_launch(void* const* d_in, const int* in_sizes, int n_in,
                              void* d_out, int out_size, void* d_ws, size_t ws_size,
                              hipStream_t stream) {
    const float* x       = (const float*)d_in[0];
    const int*   ei      = (const int*)d_in[1];
    const float* Wf_rel  = (const float*)d_in[2];
    const float* Wf_root = (const float*)d_in[3];
    const float* bf      = (const float*)d_in[4];
    const float* Ws_rel  = (const float*)d_in[5];
    const float* Ws_root = (const float*)d_in[6];
    const float* bs      = (const float*)d_in[7];
    const float* Wp_rel  = (const float*)d_in[8];
    const float* Wp_root = (const float*)d_in[9];
    const float* bp      = (const float*)d_in[10];

    const int n_nodes = in_sizes[0] / 3;       // 100000
    const int n_edges = in_sizes[1] / 2;       // 600000
    const int* src = ei;
    const int* dst = ei + n_edges;

    const int feat_elems = n_nodes * HID;
    const int wtotal = NLAYERS * HID * HID;

    float* hA    = (float*)d_ws;                        // ping buffer
    float* agg   = hA + (size_t)feat_elems;             // aggregation scratch
    float* agg3  = agg + (size_t)feat_elems;            // conv_first scratch
    float* packR = agg3 + (size_t)n_nodes * 3;          // packed Ws_rel
    float* packT = packR + (size_t)wtotal;              // packed Ws_root
    float* feat  = (float*)d_out;                       // pong buffer == out_features
    float* pos   = feat + (size_t)feat_elems;           // out_pos

    // ---- one-time per call: pack weights, zero accumulators ----
    pack_weights_kernel<<<(wtotal + 255) / 256, 256, 0, stream>>>(Ws_rel, packR, wtotal);
    pack_weights_kernel<<<(wtotal + 255) / 256, 256, 0, stream>>>(Ws_root, packT, wtotal);
    zero_kernel<<<(n_nodes * 3 + 255) / 256, 256, 0, stream>>>(agg3, n_nodes * 3);
    zero_kernel<<<(feat_elems + 255) / 256, 256, 0, stream>>>(agg, feat_elems);

    // ---- conv_first (3 -> 128) ----
    scatter3_kernel<<<(n_edges + 255) / 256, 256, 0, stream>>>(x, src, dst, agg3, n_edges);
    conv_first_kernel<<<(feat_elems + 255) / 256, 256, 0, stream>>>(
        agg3, x, Wf_rel, Wf_root, bf, hA, n_nodes);

    // ---- 11 residual GraphConv layers (128 -> 128) ----
    // gconv_gemm re-zeroes agg each layer, so no zero pass in the loop.
    float* cur = hA;
    float* nxt = feat;
    for (int l = 0; l < NLAYERS; ++l) {
        scatter128_kernel<<<(n_edges * 32 + 255) / 256, 256, 0, stream>>>(
            cur, src, dst, agg, n_edges);
        gconv_gemm_kernel<<<n_nodes / 32, 256, 0, stream>>>(
            agg, cur,
            packR + (size_t)l * HID * HID,
            packT + (size_t)l * HID * HID,
            bs + (size_t)l * HID,
            nxt);
        float* t = cur; cur = nxt; nxt = t;
    }
    // 11 layers (odd) -> final features already in d_out (feat). Safety copy if not.
    if (cur != feat) {
        hipMemcpyAsync(feat, cur, (size_t)feat_elems * sizeof(float),
                       hipMemcpyDeviceToDevice, stream);
        cur = feat;
    }

    // ---- conv_pos (128 -> 3); agg was re-zeroed by the last gemm ----
    scatter128_kernel<<<(n_edges * 32 + 255) / 256, 256, 0, stream>>>(
        cur, src, dst, agg, n_edges);
    conv_pos_kernel<<<(n_nodes * 3 + 255) / 256, 256, 0, stream>>>(
        agg, cur, Wp_rel, Wp_root, bp, pos, n_nodes);
}